// Voxel_RefinerXL_25451976196866
// MI455X (gfx1250) — compile-verified
//
#include <hip/hip_runtime.h>

typedef __attribute__((ext_vector_type(2))) float v2f;
typedef __attribute__((ext_vector_type(8))) float v8f;

#define DIMX  128
#define NVOX  (128*128*128)
#define KTOT  216           // 8 input channels * 27 taps
#define XS_W  130           // 128 + 2 halo
#define XS_SZ (8*3*3*XS_W)  // 9360 floats = 37440 B

// im2col LDS offset for flat K index k = ic*27 + t (constant-folds under full unroll)
__device__ __forceinline__ int xoff(int k) {
    const int ic = k / 27, t = k % 27;
    return ((ic * 3 + t / 9) * 3 + (t / 3) % 3) * XS_W + (t % 3);
}

// Stage an 8ch x 3d x 3h x 130w halo tile for row (d0,h0) into LDS (zero-padded).
__device__ __forceinline__ void stage_tile(const float* __restrict__ src, float* xs,
                                           int d0, int h0, int tid, int nthr) {
    for (int i = tid; i < XS_SZ; i += nthr) {
        int col  = i % XS_W;
        int rest = i / XS_W;
        int dh = rest % 3; rest /= 3;
        int dd = rest % 3;
        int ic = rest / 3;
        int z = d0 + dd - 1;
        int y = h0 + dh - 1;
        int w = col - 1;
        float v = 0.0f;
        if ((unsigned)z < 128u && (unsigned)y < 128u && (unsigned)w < 128u)
            v = src[ic * NVOX + (z * DIMX + y) * DIMX + w];
        xs[i] = v;
    }
}

// ---------------- conv1 (8->8, 3x3x3) + bias + ReLU, via V_WMMA_F32_16X16X4_F32
__global__ __launch_bounds__(256) void conv1_relu_wmma(const float* __restrict__ x,
                                                       const float* __restrict__ w1,
                                                       const float* __restrict__ b1,
                                                       float* __restrict__ h) {
    __shared__ __align__(16) float xs[XS_SZ];
    // K-pair interleaved weights: As[(k>>1)*32 + m*2 + (k&1)]  (rows m>=8 zero)
    __shared__ __align__(16) float As[KTOT * 16];
    const int tid = threadIdx.x;
    const int d0  = blockIdx.x >> 7;
    const int h0  = blockIdx.x & 127;

    stage_tile(x, xs, d0, h0, tid, 256);
    for (int i = tid; i < KTOT * 16; i += 256) {
        const int m = i & 15, k = i >> 4;
        As[(k >> 1) * 32 + m * 2 + (k & 1)] = (m < 8) ? w1[m * KTOT + k] : 0.0f;
    }
    __syncthreads();

    const int lane = tid & 31;
    const int wave = tid >> 5;
    const int half = lane >> 4;      // which K half of the fragment this lane holds
    const int m    = lane & 15;
    const int col  = wave * 16 + m;  // voxel column (N) within this row

    const float* Ap = As + (half * 16 + m) * 2;  // per-lane A base (pair layout)
    const float* Xc = xs + col;                  // per-lane B base

    v8f c = {0.f, 0.f, 0.f, 0.f, 0.f, 0.f, 0.f, 0.f};
    #pragma unroll
    for (int k0 = 0; k0 < KTOT; k0 += 4) {
        // A: one aligned b64 load at compile-time offset
        const v2f a = *(const v2f*)(Ap + (k0 >> 1) * 32);
        // B: two b32 loads; offsets are literals selected by `half`
        const int o0 = half ? xoff(k0 + 2) : xoff(k0 + 0);
        const int o1 = half ? xoff(k0 + 3) : xoff(k0 + 1);
        v2f b;
        b.x = Xc[o0];
        b.y = Xc[o1];
        c = __builtin_amdgcn_wmma_f32_16x16x4_f32(false, a, false, b,
                                                  (short)0, c, false, false);
    }

    if (lane < 16) {  // lanes>=16 hold padded rows M=8..15 (zero weights)
        const int p = (d0 * DIMX + h0) * DIMX + col;
        #pragma unroll
        for (int r = 0; r < 8; ++r)
            h[r * NVOX + p] = fmaxf(c[r] + b1[r], 0.0f);
    }
}

// ---------------- conv2 (8->27, 3x3x3) + L1 channel normalize
__global__ __launch_bounds__(256) void conv2_norm_wmma(const float* __restrict__ hsrc,
                                                       const float* __restrict__ w2,
                                                       float* __restrict__ wout) {
    __shared__ __align__(16) float xs[XS_SZ];
    // K-pair interleaved, two 16-row M-tiles: As[(k>>1)*64 + m*2 + (k&1)], m<27 valid
    __shared__ __align__(16) float As[KTOT * 32];
    const int tid = threadIdx.x;
    const int d0  = blockIdx.x >> 7;
    const int h0  = blockIdx.x & 127;

    stage_tile(hsrc, xs, d0, h0, tid, 256);
    for (int i = tid; i < KTOT * 32; i += 256) {
        const int m = i & 31, k = i >> 5;
        As[(k >> 1) * 64 + m * 2 + (k & 1)] = (m < 27) ? w2[m * KTOT + k] : 0.0f;
    }
    __syncthreads();

    const int lane = tid & 31;
    const int wave = tid >> 5;
    const int half = lane >> 4;
    const int m    = lane & 15;
    const int col  = wave * 16 + m;

    const float* Ap0 = As + (half * 32 + m) * 2;        // tile0: oc 0..15
    const float* Ap1 = Ap0 + 32;                        // tile1: oc 16..31
    const float* Xc  = xs + col;

    v8f c0 = {0.f, 0.f, 0.f, 0.f, 0.f, 0.f, 0.f, 0.f};
    v8f c1 = {0.f, 0.f, 0.f, 0.f, 0.f, 0.f, 0.f, 0.f};
    #pragma unroll
    for (int k0 = 0; k0 < KTOT; k0 += 4) {
        const v2f a0 = *(const v2f*)(Ap0 + (k0 >> 1) * 64);
        const v2f a1 = *(const v2f*)(Ap1 + (k0 >> 1) * 64);
        const int o0 = half ? xoff(k0 + 2) : xoff(k0 + 0);
        const int o1 = half ? xoff(k0 + 3) : xoff(k0 + 1);
        v2f b;
        b.x = Xc[o0];
        b.y = Xc[o1];
        c0 = __builtin_amdgcn_wmma_f32_16x16x4_f32(false, a0, false, b,
                                                   (short)0, c0, false, false);
        c1 = __builtin_amdgcn_wmma_f32_16x16x4_f32(false, a1, false, b,
                                                   (short)0, c1, false, false);
    }

    // L1 norm over the 27 channels of this voxel (split across lane and lane^16)
    float s = 0.0f;
    #pragma unroll
    for (int r = 0; r < 8; ++r) s += fabsf(c0[r]);        // oc 0..7 / 8..15
    if (lane < 16) {
        #pragma unroll
        for (int r = 0; r < 8; ++r) s += fabsf(c1[r]);    // oc 16..23
    } else {
        s += fabsf(c1[0]) + fabsf(c1[1]) + fabsf(c1[2]);  // oc 24..26
    }
    s += __shfl_xor(s, 16, 32);                            // combine wave32 halves
    const float inv = 1.0f / fmaxf(s, 1e-12f);

    const int p = (d0 * DIMX + h0) * DIMX + col;
    if (lane < 16) {
        #pragma unroll
        for (int r = 0; r < 8; ++r) wout[r * NVOX + p]        = c0[r] * inv;
        #pragma unroll
        for (int r = 0; r < 8; ++r) wout[(16 + r) * NVOX + p] = c1[r] * inv;
    } else {
        #pragma unroll
        for (int r = 0; r < 8; ++r) wout[(8 + r) * NVOX + p]  = c0[r] * inv;
        #pragma unroll
        for (int r = 0; r < 3; ++r) wout[(24 + r) * NVOX + p] = c1[r] * inv;
    }
}

// ---------------- one adaptive-conv pass (bandwidth-bound; LDS-tiled VALU)
__global__ __launch_bounds__(128) void adaptive_apply(const float* __restrict__ in,
                                                      const float* __restrict__ wv,
                                                      float* __restrict__ out) {
    __shared__ __align__(16) float xs[XS_SZ];
    const int tid = threadIdx.x;
    const int d0  = blockIdx.x >> 7;
    const int h0  = blockIdx.x & 127;

    stage_tile(in, xs, d0, h0, tid, 128);
    __syncthreads();

    const int p = (d0 * DIMX + h0) * DIMX + tid;
    float acc[8] = {0.f, 0.f, 0.f, 0.f, 0.f, 0.f, 0.f, 0.f};
    #pragma unroll
    for (int t = 0; t < 27; ++t) {
        const float wt = wv[t * NVOX + p];
        const int base = ((t / 9) * 3 + (t / 3) % 3) * XS_W + tid + (t % 3);
        #pragma unroll
        for (int cc = 0; cc < 8; ++cc)
            acc[cc] = fmaf(wt, xs[cc * (9 * XS_W) + base], acc[cc]);
    }
    #pragma unroll
    for (int cc = 0; cc < 8; ++cc)
        out[cc * NVOX + p] = acc[cc];
}

extern "C" void kernel_launch(void* const* d_in, const int* in_sizes, int n_in,
                              void* d_out, int out_size, void* d_ws, size_t ws_size,
                              hipStream_t stream) {
    (void)in_sizes; (void)n_in; (void)out_size; (void)ws_size;
    const float* x  = (const float*)d_in[0];
    const float* w1 = (const float*)d_in[1];
    const float* b1 = (const float*)d_in[2];
    const float* w2 = (const float*)d_in[3];
    float* out  = (float*)d_out;
    float* hbuf = (float*)d_ws;                   //  8 * NVOX floats (64 MB)
    float* w27  = hbuf + (size_t)8 * NVOX;        // 27 * NVOX floats (226 MB)

    const dim3 grid(DIMX * DIMX);                 // one block per (d,h) row
    conv1_relu_wmma<<<grid, 256, 0, stream>>>(x, w1, b1, hbuf);
    conv2_norm_wmma<<<grid, 256, 0, stream>>>(hbuf, w2, w27);
    adaptive_apply<<<grid, 128, 0, stream>>>(x,    w27, out);   // iter 1
    adaptive_apply<<<grid, 128, 0, stream>>>(out,  w27, hbuf);  // iter 2
    adaptive_apply<<<grid, 128, 0, stream>>>(hbuf, w27, out);   // iter 3
}